// MambaStack_40853728920180
// MI455X (gfx1250) — compile-verified
//
#include <hip/hip_runtime.h>
#include <math.h>

typedef float v2f __attribute__((ext_vector_type(2)));
typedef float v8f __attribute__((ext_vector_type(8)));

#define D_IN     80
#define D_MODEL  256
#define N_LAYERS 4
#define D_INNER  512
#define D_STATE  16
#define D_CONVW  4
#define DT_RANK  16
#define BB       8
#define LL       1024
#define MROWS    (BB*LL)   // 8192

// ---------------------------------------------------------------------------
// Templated WMMA K-loop: NT = number of valid 16-wide N tiles (1..4).
// Wave tile = 32(M) x 16*NT(N): 2 A fragments, NT B fragments, 2*NT WMMAs
// per K-step of 4. No EXEC manipulation inside the loop (NT is wave-uniform).
// ---------------------------------------------------------------------------
template <int NT>
__device__ __forceinline__ void gemm_kloop(const float* __restrict__ pa0,
                                           const float* __restrict__ pa1,
                                           const float* __restrict__ pb,
                                           size_t ldb16,   // 16*K
                                           int K, v8f acc[2][4])
{
    for (int k0 = 0; k0 < K; k0 += 4) {
        v2f a0 = *(const v2f*)(pa0 + k0);
        v2f a1 = *(const v2f*)(pa1 + k0);
        v2f b[NT];
        #pragma unroll
        for (int j = 0; j < NT; ++j)
            b[j] = *(const v2f*)(pb + (size_t)j * ldb16 + k0);
        #pragma unroll
        for (int j = 0; j < NT; ++j) {
            acc[0][j] = __builtin_amdgcn_wmma_f32_16x16x4_f32(
                false, a0, false, b[j], (short)0, acc[0][j], false, false);
            acc[1][j] = __builtin_amdgcn_wmma_f32_16x16x4_f32(
                false, a1, false, b[j], (short)0, acc[1][j], false, false);
        }
    }
}

// ---------------------------------------------------------------------------
// f32 WMMA GEMM:  Out[m,n] = epilogue( sum_k X[m*ldx + k] * W[n*K + k] )
// EPI: 0 = plain, 1 = +bias[n], 2 = softplus(acc + bias[n]), 3 = Out += scale*acc
// Block = 8 waves, wave tile 32M x 64N -> block tile 64M x 256N.
// ---------------------------------------------------------------------------
template <int EPI>
__global__ __launch_bounds__(256) void gemm_f32_wmma(
    const float* __restrict__ X, int ldx,
    const float* __restrict__ W,
    const float* __restrict__ bias,
    float* __restrict__ Out,
    int M, int N, int K,
    const float* __restrict__ scale_ptr)
{
    const int tid  = threadIdx.x;
    const int lane = tid & 31;
    const int wave = tid >> 5;
    const int half = lane >> 4;   // 0: lanes 0-15, 1: lanes 16-31
    const int idx  = lane & 15;
    const int wm   = wave >> 2;   // 0..1
    const int wn   = wave & 3;    // 0..3
    const int m0   = blockIdx.y * 64  + wm * 32;
    const int n0   = blockIdx.x * 256 + wn * 64;

    // wave-uniform count of valid 16-wide N tiles for this wave (N % 16 == 0)
    int nvt = (N - n0) / 16;
    if (nvt <= 0) return;
    if (nvt > 4) nvt = 4;

    v8f acc[2][4] = {};

    const int ko = 2 * half;   // per-lane K sub-offset inside a K=4 step
    const float* pa0 = X + (size_t)(m0 + idx) * ldx + ko;
    const float* pa1 = pa0 + (size_t)16 * ldx;
    const float* pb  = W + (size_t)(n0 + idx) * K + ko;
    const size_t ldb16 = (size_t)16 * K;

    switch (nvt) {
        case 4:  gemm_kloop<4>(pa0, pa1, pb, ldb16, K, acc); break;
        case 3:  gemm_kloop<3>(pa0, pa1, pb, ldb16, K, acc); break;
        case 2:  gemm_kloop<2>(pa0, pa1, pb, ldb16, K, acc); break;
        default: gemm_kloop<1>(pa0, pa1, pb, ldb16, K, acc); break;
    }

    const float sc = (EPI == 3) ? scale_ptr[0] : 0.0f;

    #pragma unroll
    for (int j = 0; j < 4; ++j) {
        if (j >= nvt) break;
        const int col = n0 + 16 * j + idx;
        #pragma unroll
        for (int i = 0; i < 2; ++i) {
            #pragma unroll
            for (int v = 0; v < 8; ++v) {
                const int row = m0 + 16 * i + v + 8 * half;  // C/D VGPR layout
                float val = acc[i][j][v];
                const size_t o = (size_t)row * N + col;
                if (EPI == 1) {
                    Out[o] = val + bias[col];
                } else if (EPI == 2) {
                    val += bias[col];
                    Out[o] = (val > 20.0f) ? val : log1pf(__expf(val));
                } else if (EPI == 3) {
                    Out[o] += sc * val;
                } else {
                    Out[o] = val;
                }
            }
        }
    }
}

// ---------------------------------------------------------------------------
// Depthwise causal conv (width 4) + bias + SiLU.  xi = xz[..., :512]
// ---------------------------------------------------------------------------
__global__ void conv_silu_kernel(const float* __restrict__ xz,
                                 const float* __restrict__ cw,
                                 const float* __restrict__ cb,
                                 float* __restrict__ xc)
{
    int g = blockIdx.x * blockDim.x + threadIdx.x;
    if (g >= MROWS * D_INNER) return;
    int d = g % D_INNER;
    int l = (g / D_INNER) % LL;
    int b = g / (D_INNER * LL);

    float acc = cb[d];
    #pragma unroll
    for (int k = 0; k < D_CONVW; ++k) {
        int ls = l + k - (D_CONVW - 1);
        if (ls >= 0)
            acc += xz[((size_t)b * LL + ls) * (2 * D_INNER) + d] * cw[d * D_CONVW + k];
    }
    float sig = 1.0f / (1.0f + __expf(-acc));
    xc[g] = acc * sig;
}

// ---------------------------------------------------------------------------
// Selective scan: block = 128 channels of one batch; 16 states/thread in regs.
// B_t / C_t staged through double-buffered LDS -> one barrier per timestep.
// Prefetch (global_prefetch_b8) pulls t+16 rows toward the WGP.
// ---------------------------------------------------------------------------
__global__ __launch_bounds__(128) void scan_kernel(
    const float* __restrict__ delta,   // (B,L,512) already softplus'd
    const float* __restrict__ dbc,     // (B,L,48)
    const float* __restrict__ xc,      // (B,L,512)
    const float* __restrict__ xz,      // (B,L,1024); z = [...,512:]
    const float* __restrict__ A_log,   // (512,16) for this layer
    const float* __restrict__ Dvec,    // (512)
    float* __restrict__ ybuf)          // (B,L,512)
{
    const int tid = threadIdx.x;
    const int d   = blockIdx.x * 128 + tid;
    const int b   = blockIdx.y;

    float A[D_STATE];
    #pragma unroll
    for (int n = 0; n < D_STATE; ++n) A[n] = -__expf(A_log[d * D_STATE + n]);
    const float Dd = Dvec[d];

    float h[D_STATE];
    #pragma unroll
    for (int n = 0; n < D_STATE; ++n) h[n] = 0.0f;

    __shared__ float sBC[2][2 * D_STATE];

    const size_t rowbase = (size_t)b * LL;
    if (tid < 2 * D_STATE) sBC[0][tid] = dbc[rowbase * 48 + DT_RANK + tid];
    __syncthreads();

    for (int t = 0; t < LL; ++t) {
        const int cur = t & 1;
        const size_t r = rowbase + t;

        // stage next timestep's B/C into the other buffer (no extra barrier)
        if (tid < 2 * D_STATE && (t + 1) < LL)
            sBC[cur ^ 1][tid] = dbc[(r + 1) * 48 + DT_RANK + tid];

        // prefetch ahead of the dependent chain
        if ((t + 16) < LL) {
            __builtin_prefetch(&delta[(r + 16) * D_INNER + d], 0, 1);
            __builtin_prefetch(&xc[(r + 16) * D_INNER + d], 0, 1);
            __builtin_prefetch(&xz[(r + 16) * (2 * D_INNER) + D_INNER + d], 0, 1);
        }

        const float dt = delta[r * D_INNER + d];
        const float u  = xc[r * D_INNER + d];
        const float zv = xz[r * (2 * D_INNER) + D_INNER + d];
        const float du = dt * u;

        float y = 0.0f;
        #pragma unroll
        for (int n = 0; n < D_STATE; ++n) {
            float dA = __expf(dt * A[n]);
            h[n] = h[n] * dA + du * sBC[cur][n];
            y += h[n] * sBC[cur][D_STATE + n];
        }
        float sig = 1.0f / (1.0f + __expf(-zv));
        ybuf[r * D_INNER + d] = (y + u * Dd) * (zv * sig);
        __syncthreads();
    }
}

// ---------------------------------------------------------------------------
// LayerNorm over D_MODEL=256: one wave per row, 8 elems/lane, shfl_xor reduce.
// ---------------------------------------------------------------------------
__global__ __launch_bounds__(256) void ln_kernel(
    const float* __restrict__ in,
    const float* __restrict__ gamma,
    const float* __restrict__ beta,
    float* __restrict__ out, int rows)
{
    const int lane = threadIdx.x & 31;
    const int wave = threadIdx.x >> 5;
    const int row  = blockIdx.x * 8 + wave;
    if (row >= rows) return;

    const float* rp = in + (size_t)row * D_MODEL;
    float v[8];
    float s = 0.0f;
    #pragma unroll
    for (int j = 0; j < 8; ++j) { v[j] = rp[j * 32 + lane]; s += v[j]; }
    #pragma unroll
    for (int m = 16; m >= 1; m >>= 1) s += __shfl_xor(s, m, 32);
    const float mu = s * (1.0f / D_MODEL);

    float q = 0.0f;
    #pragma unroll
    for (int j = 0; j < 8; ++j) { float dd = v[j] - mu; q += dd * dd; }
    #pragma unroll
    for (int m = 16; m >= 1; m >>= 1) q += __shfl_xor(q, m, 32);
    const float inv = rsqrtf(q * (1.0f / D_MODEL) + 1e-5f);

    float* op = out + (size_t)row * D_MODEL;
    #pragma unroll
    for (int j = 0; j < 8; ++j) {
        int c = j * 32 + lane;
        op[c] = (v[j] - mu) * inv * gamma[c] + beta[c];
    }
}

// ---------------------------------------------------------------------------
extern "C" void kernel_launch(void* const* d_in, const int* in_sizes, int n_in,
                              void* d_out, int out_size, void* d_ws, size_t ws_size,
                              hipStream_t stream)
{
    (void)in_sizes; (void)n_in; (void)out_size; (void)ws_size;

    const float* x          = (const float*)d_in[0];
    const float* proj_in_w  = (const float*)d_in[1];
    const float* proj_in_b  = (const float*)d_in[2];
    const float* ln_gamma   = (const float*)d_in[3];
    const float* ln_beta    = (const float*)d_in[4];
    const float* in_proj_w  = (const float*)d_in[5];
    const float* conv_w     = (const float*)d_in[6];
    const float* conv_b     = (const float*)d_in[7];
    const float* x_proj_w   = (const float*)d_in[8];
    const float* dt_proj_w  = (const float*)d_in[9];
    const float* dt_proj_b  = (const float*)d_in[10];
    const float* A_log      = (const float*)d_in[11];
    const float* Dw         = (const float*)d_in[12];
    const float* out_proj_w = (const float*)d_in[13];
    const float* res_scale  = (const float*)d_in[14];
    const float* on_gamma   = (const float*)d_in[15];
    const float* on_beta    = (const float*)d_in[16];

    float* ws = (float*)d_ws;
    size_t o = 0;
    float* h     = ws + o; o += (size_t)MROWS * D_MODEL;        // 8 MB
    float* lnb   = ws + o; o += (size_t)MROWS * D_MODEL;        // 8 MB
    float* xz    = ws + o; o += (size_t)MROWS * 2 * D_INNER;    // 32 MB
    float* xc    = ws + o; o += (size_t)MROWS * D_INNER;        // 16 MB
    float* dbc   = ws + o; o += (size_t)MROWS * 48;             // 1.5 MB
    float* delta = ws + o; o += (size_t)MROWS * D_INNER;        // 16 MB
    float* ybuf  = ws + o;                                      // 16 MB

    const dim3 blk(256);

    // h = x @ proj_in_w.T + proj_in_b   (M=8192, N=256, K=80)
    gemm_f32_wmma<1><<<dim3(1, 128), blk, 0, stream>>>(
        x, D_IN, proj_in_w, proj_in_b, h, MROWS, D_MODEL, D_IN, nullptr);

    for (int i = 0; i < N_LAYERS; ++i) {
        // LayerNorm(h) -> lnb
        ln_kernel<<<MROWS / 8, 256, 0, stream>>>(
            h, ln_gamma + i * D_MODEL, ln_beta + i * D_MODEL, lnb, MROWS);

        // xz = lnb @ in_proj_w[i].T   (N=1024, K=256)
        gemm_f32_wmma<0><<<dim3(4, 128), blk, 0, stream>>>(
            lnb, D_MODEL, in_proj_w + (size_t)i * 2 * D_INNER * D_MODEL,
            nullptr, xz, MROWS, 2 * D_INNER, D_MODEL, nullptr);

        // xc = silu(causal_conv4(xz[...,:512]) + conv_b)
        conv_silu_kernel<<<(MROWS * D_INNER) / 256, 256, 0, stream>>>(
            xz, conv_w + (size_t)i * D_INNER * D_CONVW, conv_b + i * D_INNER, xc);

        // dbc = xc @ x_proj_w[i].T   (N=48, K=512)
        gemm_f32_wmma<0><<<dim3(1, 128), blk, 0, stream>>>(
            xc, D_INNER, x_proj_w + (size_t)i * 48 * D_INNER,
            nullptr, dbc, MROWS, 48, D_INNER, nullptr);

        // delta = softplus(dbc[:, :16] @ dt_proj_w[i].T + dt_proj_b[i])  (N=512, K=16, ldx=48)
        gemm_f32_wmma<2><<<dim3(2, 128), blk, 0, stream>>>(
            dbc, 48, dt_proj_w + (size_t)i * D_INNER * DT_RANK,
            dt_proj_b + i * D_INNER, delta, MROWS, D_INNER, DT_RANK, nullptr);

        // selective scan -> ybuf
        scan_kernel<<<dim3(D_INNER / 128, BB), 128, 0, stream>>>(
            delta, dbc, xc, xz, A_log + (size_t)i * D_INNER * D_STATE,
            Dw + i * D_INNER, ybuf);

        // h += res_scale * (ybuf @ out_proj_w[i].T)   (N=256, K=512)
        gemm_f32_wmma<3><<<dim3(1, 128), blk, 0, stream>>>(
            ybuf, D_INNER, out_proj_w + (size_t)i * D_MODEL * D_INNER,
            nullptr, h, MROWS, D_MODEL, D_INNER, res_scale);
    }

    // final LayerNorm -> d_out
    ln_kernel<<<MROWS / 8, 256, 0, stream>>>(h, on_gamma, on_beta, (float*)d_out, MROWS);
}